// Quantizer_3264175145006
// MI455X (gfx1250) — compile-verified
//
#include <hip/hip_runtime.h>
#include <hip/hip_bf16.h>
#include <stdint.h>

#define DECAY 0.9f
#define ONE_MINUS_DECAY 0.1f
#define EPSV 1e-5f

#define NROWS 131072   /* 32*64*64 */
#define DDIM 64
#define KCODES 1024

#define CBLK 128        /* codebook columns staged per LDS pass            */
#define LDS_STRIDE 68   /* floats per staged column: 64 data + 4 pad       */
                        /* 68*4 = 272 B (8B aligned), bank step 4 ->       */
                        /* 32-lane ds_load_b64 across 16 cols x 2 halves   */
                        /* touches all 64 banks exactly once               */

typedef __attribute__((ext_vector_type(2))) float v2f;
typedef __attribute__((ext_vector_type(8))) float v8f;

/* ---- workspace layout (bytes) ---- */
#define WS_IDX_OFF    0                              /* int[NROWS]          */
#define WS_COUNTS_OFF (NROWS * 4)                    /* float[KCODES]       */
#define WS_ISUM_OFF   (WS_COUNTS_OFF + KCODES * 4)   /* float[DDIM*KCODES]  */
#define WS_C2_OFF     (WS_ISUM_OFF + DDIM * KCODES * 4) /* float[KCODES]    */
#define WS_N_OFF      (WS_C2_OFF + KCODES * 4)       /* float[1]            */

/* ---- output layout (floats), tuple order of the reference ---- */
#define OUT_Q_OFF    0
#define OUT_LOSS_OFF (NROWS * DDIM)
#define OUT_PERP_OFF (OUT_LOSS_OFF + 1)
#define OUT_CB_OFF   (OUT_PERP_OFF + 1)
#define OUT_CS_OFF   (OUT_CB_OFF + DDIM * KCODES)
#define OUT_SUM_OFF  (OUT_CS_OFF + KCODES)

/* ------------------------------------------------------------------ */
__global__ void k_zero(float* counts, float* isum, float* loss_slot) {
    int tid = blockIdx.x * blockDim.x + threadIdx.x;
    int total = KCODES + DDIM * KCODES;
    for (int i = tid; i < total; i += gridDim.x * blockDim.x) {
        if (i < KCODES) counts[i] = 0.0f;
        else            isum[i - KCODES] = 0.0f;
    }
    if (tid == 0) loss_slot[0] = 0.0f;
}

/* ------------------------------------------------------------------ */
__global__ void k_c2(const float* __restrict__ cb, float* __restrict__ c2) {
    int k = blockIdx.x * blockDim.x + threadIdx.x;
    if (k < KCODES) {
        float s = 0.0f;
#pragma unroll
        for (int d = 0; d < DDIM; ++d) {
            float v = cb[d * KCODES + k];
            s = __builtin_fmaf(v, v, s);
        }
        c2[k] = s;
    }
}

/* ------------------------------------------------------------------ */
/* main: f32 WMMA distance GEMM + argmin + scatter.                   */
/* Block = 8 waves; wave owns 32 rows (two 16x64 A tiles resident).   */
/* Codebook staged in LDS in 128-col transposed blocks; B fragment    */
/* per lane = contiguous 8B -> ds_load_2addr_b64, conflict-free.      */
/* B tile preloaded into a register array so the DS clause can stay   */
/* deep while the WMMA chains retire.                                 */
__global__ __launch_bounds__(256) void k_assign(
        const float* __restrict__ flat,
        const float* __restrict__ cb,
        const float* __restrict__ c2,
        int*   __restrict__ idxArr,
        float* __restrict__ counts,
        float* __restrict__ isum) {
    __shared__ float scb[CBLK * LDS_STRIDE];       /* 34816 B */
    __shared__ __align__(16) float sc2[CBLK];      /*   512 B */

    const int lane = threadIdx.x & 31;
    const int wave = threadIdx.x >> 5;
    const int rowBase = (blockIdx.x * (blockDim.x >> 5) + wave) * 32;
    const int mrow = lane & 15;
    const int half = lane >> 4;

    /* A tiles resident: chunk c covers K = 4c..4c+3.
       ISA 7.12.2 (32-bit A 16x4): VGPR0 = K0|K2, VGPR1 = K1|K3
       -> lane holds the contiguous pair at K = 4c + 2*half.  */
    const float* arow0 = flat + (size_t)(rowBase + mrow) * DDIM;
    const float* arow1 = flat + (size_t)(rowBase + 16 + mrow) * DDIM;
    v2f a0[16], a1[16];
#pragma unroll
    for (int c = 0; c < 16; ++c) {
        a0[c] = *(const v2f*)(arow0 + 4 * c + 2 * half);
        a1[c] = *(const v2f*)(arow1 + 4 * c + 2 * half);
    }

    float best[16];
    int   bidx[16];
#pragma unroll
    for (int s = 0; s < 16; ++s) { best[s] = 3.4e38f; bidx[s] = 0; }

    for (int blk = 0; blk < KCODES / CBLK; ++blk) {
        const int colBase = blk * CBLK;

        /* stage transposed codebook block: scb[c * 68 + d] = cb[d, colBase+c].
           Global side coalesced (128 consecutive k per half-block of threads). */
#pragma unroll 4
        for (int i = 0; i < (DDIM * CBLK) / 256; ++i) {
            int e = i * 256 + threadIdx.x;
            int d = e >> 7;
            int cc = e & (CBLK - 1);
            scb[cc * LDS_STRIDE + d] = cb[d * KCODES + colBase + cc];
        }
        /* async-copy the ||c||^2 slice for this block: 512 B via one
           global_load_async_to_lds_b128 issued by wave 0 (ASYNCcnt). */
        if (wave == 0) {
            unsigned ldsa = (unsigned)(uintptr_t)(&sc2[0]) + lane * 16u;
            const float* gsrc = c2 + colBase + lane * 4;
            asm volatile("global_load_async_to_lds_b128 %0, %1, off"
                         :: "v"(ldsa), "v"(gsrc) : "memory");
            asm volatile("s_wait_asynccnt 0" ::: "memory");
        }
        __syncthreads();

        for (int t = 0; t < CBLK / 16; ++t) {
            const int colL = t * 16 + mrow;
            const float* bp = scb + colL * LDS_STRIDE + 2 * half;

            /* preload whole B tile: 8x ds_load_2addr_b64 issued as one
               clause; waits then drain progressively under the WMMAs */
            v2f bfr[16];
#pragma unroll
            for (int c = 0; c < 16; ++c)
                bfr[c] = *(const v2f*)(bp + 4 * c);

            v8f acc0 = {}, acc1 = {};
#pragma unroll
            for (int c = 0; c < 16; ++c) {
                /* B 4x16, rows striped across lanes (mirrors A):
                   lane pair = rows (4c+2*half, 4c+1+2*half) at one col */
                acc0 = __builtin_amdgcn_wmma_f32_16x16x4_f32(
                    false, a0[c], false, bfr[c], (short)0, acc0, false, false);
                acc1 = __builtin_amdgcn_wmma_f32_16x16x4_f32(
                    false, a1[c], false, bfr[c], (short)0, acc1, false, false);
            }
            const float c2v = sc2[colL];
            const int col = colBase + colL;
#pragma unroll
            for (int r = 0; r < 8; ++r) {
                float d0 = __builtin_fmaf(-2.0f, acc0[r], c2v);
                float d1 = __builtin_fmaf(-2.0f, acc1[r], c2v);
                if (d0 < best[r])     { best[r] = d0;     bidx[r] = col; }
                if (d1 < best[8 + r]) { best[8 + r] = d1; bidx[8 + r] = col; }
            }
        }
        __syncthreads();
    }

    /* cross-lane argmin over the 16 lanes sharing each row set;
       tie-break to the lowest code index (jnp.argmin semantics) */
#pragma unroll
    for (int off = 1; off < 16; off <<= 1) {
#pragma unroll
        for (int s = 0; s < 16; ++s) {
            float ov = __shfl_xor(best[s], off, 32);
            int   oi = __shfl_xor(bidx[s], off, 32);
            if (ov < best[s] || (ov == best[s] && oi < bidx[s])) {
                best[s] = ov; bidx[s] = oi;
            }
        }
    }

    /* slot s = m*8 + r holds: lane0 -> row rowBase + 16m + r,
                               lane16 -> row rowBase + 16m + 8 + r */
#pragma unroll
    for (int m = 0; m < 2; ++m) {
#pragma unroll
        for (int r = 0; r < 8; ++r) {
            const int s = m * 8 + r;
            int k0 = __shfl(bidx[s], 0, 32);
            int k1 = __shfl(bidx[s], 16, 32);
            const int row0 = rowBase + 16 * m + r;
            const int row1 = rowBase + 16 * m + 8 + r;
            if (lane == 0) {
                idxArr[row0] = k0;
                unsafeAtomicAdd(&counts[k0], 1.0f);
            }
            if (lane == 16) {
                idxArr[row1] = k1;
                unsafeAtomicAdd(&counts[k1], 1.0f);
            }
            const float* r0 = flat + (size_t)row0 * DDIM;
            const float* r1 = flat + (size_t)row1 * DDIM;
            unsafeAtomicAdd(&isum[(size_t)lane        * KCODES + k0], r0[lane]);
            unsafeAtomicAdd(&isum[(size_t)(lane + 32) * KCODES + k0], r0[lane + 32]);
            unsafeAtomicAdd(&isum[(size_t)lane        * KCODES + k1], r1[lane]);
            unsafeAtomicAdd(&isum[(size_t)(lane + 32) * KCODES + k1], r1[lane + 32]);
        }
    }
}

/* ------------------------------------------------------------------ */
__global__ __launch_bounds__(1024) void k_ema_cs(
        const float* __restrict__ counts,
        const float* __restrict__ ema_cs,
        float* __restrict__ out_cs,
        float* __restrict__ out_perp,
        float* __restrict__ n_out) {
    __shared__ float s1[1024];
    __shared__ float s2[1024];
    int k = threadIdx.x;
    float cnt = counts[k];
    float ncs = ema_cs[k] * DECAY + ONE_MINUS_DECAY * cnt;
    out_cs[k] = ncs;
    float p = cnt * (1.0f / (float)NROWS);
    s1[k] = ncs;
    s2[k] = -p * logf(p + 1e-10f);
    __syncthreads();
    for (int s = 512; s > 0; s >>= 1) {
        if (k < s) { s1[k] += s1[k + s]; s2[k] += s2[k + s]; }
        __syncthreads();
    }
    if (k == 0) { n_out[0] = s1[0]; out_perp[0] = expf(s2[0]); }
}

/* ------------------------------------------------------------------ */
__global__ void k_codebook(
        const float* __restrict__ ema_sum,
        const float* __restrict__ isum,
        const float* __restrict__ out_cs,
        const float* __restrict__ n_ptr,
        float* __restrict__ out_sum,
        float* __restrict__ out_cb) {
    int e = blockIdx.x * blockDim.x + threadIdx.x;
    if (e >= DDIM * KCODES) return;
    int k = e & (KCODES - 1);
    float nsum = ema_sum[e] * DECAY + ONE_MINUS_DECAY * isum[e];
    float n = n_ptr[0];
    float ncs = out_cs[k];
    float norm = (ncs + EPSV) / (n + KCODES * EPSV) * n;
    out_sum[e] = nsum;
    out_cb[e]  = nsum / norm;
}

/* ------------------------------------------------------------------ */
__global__ __launch_bounds__(256) void k_gather_loss(
        const float* __restrict__ inputs,
        const float* __restrict__ out_cb,
        const int*   __restrict__ idxArr,
        float* __restrict__ out_q,
        float* __restrict__ loss_slot) {
    const long total = (long)NROWS * DDIM;
    long tid = (long)blockIdx.x * blockDim.x + threadIdx.x;
    long stride = (long)gridDim.x * blockDim.x;
    float local = 0.0f;
    for (long ee = tid; ee < total; ee += stride) {
        int row = (int)(ee >> 6);
        int dd  = (int)(ee & 63);
        int k = idxArr[row];
        float q = out_cb[dd * KCODES + k];
        out_q[ee] = q;  /* straight-through: inputs + (q - inputs) == q */
        float diff = q - inputs[ee];
        local = __builtin_fmaf(diff, diff, local);
    }
    __shared__ float sm[8];
    for (int off = 16; off >= 1; off >>= 1) local += __shfl_xor(local, off, 32);
    int lane = threadIdx.x & 31, wv = threadIdx.x >> 5;
    if (lane == 0) sm[wv] = local;
    __syncthreads();
    if (wv == 0) {
        float v = (lane < (int)(blockDim.x >> 5)) ? sm[lane] : 0.0f;
        for (int off = 4; off >= 1; off >>= 1) v += __shfl_xor(v, off, 32);
        if (lane == 0) unsafeAtomicAdd(loss_slot, v * (1.0f / (float)total));
    }
}

/* ------------------------------------------------------------------ */
extern "C" void kernel_launch(void* const* d_in, const int* in_sizes, int n_in,
                              void* d_out, int out_size, void* d_ws, size_t ws_size,
                              hipStream_t stream) {
    const float* inputs   = (const float*)d_in[0];  /* [32,64,64,64]  */
    const float* codebook = (const float*)d_in[1];  /* [64,1024]      */
    const float* ema_cs   = (const float*)d_in[2];  /* [1024]         */
    const float* ema_sum  = (const float*)d_in[3];  /* [64,1024]      */
    float* out = (float*)d_out;
    char*  ws  = (char*)d_ws;

    int*   idxArr = (int*)  (ws + WS_IDX_OFF);
    float* counts = (float*)(ws + WS_COUNTS_OFF);
    float* isum   = (float*)(ws + WS_ISUM_OFF);
    float* c2     = (float*)(ws + WS_C2_OFF);
    float* nscal  = (float*)(ws + WS_N_OFF);

    k_zero<<<128, 256, 0, stream>>>(counts, isum, out + OUT_LOSS_OFF);
    k_c2<<<KCODES / 256, 256, 0, stream>>>(codebook, c2);
    /* 131072 rows / 32 rows-per-wave / 8 waves-per-block = 512 blocks */
    k_assign<<<NROWS / 32 / 8, 256, 0, stream>>>(inputs, codebook, c2,
                                                 idxArr, counts, isum);
    k_ema_cs<<<1, 1024, 0, stream>>>(counts, ema_cs, out + OUT_CS_OFF,
                                     out + OUT_PERP_OFF, nscal);
    k_codebook<<<(DDIM * KCODES) / 256, 256, 0, stream>>>(
        ema_sum, isum, out + OUT_CS_OFF, nscal,
        out + OUT_SUM_OFF, out + OUT_CB_OFF);
    k_gather_loss<<<512, 256, 0, stream>>>(inputs, out + OUT_CB_OFF, idxArr,
                                           out + OUT_Q_OFF, out + OUT_LOSS_OFF);
}